// PCENTransform_541165879304
// MI455X (gfx1250) — compile-verified
//
#include <hip/hip_runtime.h>

// ---------------------------------------------------------------------------
// PCEN for x:(64,1,128,4096) fp32 on gfx1250.
// - IIR scan along T recast as 16x16 lower-triangular matmuls ->
//   chained v_wmma_f32_16x16x4_f32 (4 per 16-step chunk, 32 per tile).
// - Tiles of 16 rows x 128 timesteps staged through LDS with CDNA5
//   global_load_async_to_lds_b128, double buffered, s_wait_asynccnt.
// - One wave32 per (b, 16-row m group): 512 workgroups, no barriers needed.
// - pcen elementwise: scalar-uniform fast path (alpha=r=1) with v_rcp_f32,
//   general path with v_log_f32/v_exp_f32 hardware transcendentals
//   (keeps the hot loop tiny -> no I$ thrash from inlined libm powf).
// ---------------------------------------------------------------------------

typedef float v2f __attribute__((ext_vector_type(2)));
typedef float v8f __attribute__((ext_vector_type(8)));

#define T_LEN   4096
#define M_LEN   128
#define TT      128                 // timesteps per LDS tile
#define PAD     (TT + 4)            // row stride in floats: 16B-aligned
#define ROWS    16                  // m-rows per workgroup
#define BUF_F   (ROWS * PAD)        // floats per LDS buffer (2112)
#define NTILES  (T_LEN / TT)        // 32
#define EPS_F   1e-6f

#if __has_builtin(__builtin_amdgcn_wmma_f32_16x16x4_f32)
#define HAVE_WMMA_F32X4 1
#else
#define HAVE_WMMA_F32X4 0
#endif

__device__ __forceinline__ float ipow(float a, int n) {
  float r = 1.0f;
  for (int i = 0; i < n; ++i) r *= a;   // exact for a=0, no powf edge cases
  return r;
}

__device__ __forceinline__ float pw_cold(float x, float e) {
  if (e == 1.0f) return x;              // exact (test inputs: r==1 -> dr==delta)
  return powf(x, e);                    // cold: used once outside all loops
}

// x^e via hardware v_log_f32 / v_exp_f32 (log2/exp2). ~2 ulp, 3 VALU ops.
__device__ __forceinline__ float pw_hw(float x, float e) {
  return __builtin_amdgcn_exp2f(e * __builtin_amdgcn_logf(x));
}

__device__ __forceinline__ void async_ld_b128(unsigned lds_byte, const void* g) {
  // CDNA5 async global->LDS DMA (ASYNCcnt-tracked). VDST VGPR = LDS byte addr.
  asm volatile("global_load_async_to_lds_b128 %0, %1, off"
               :: "v"(lds_byte), "v"(g) : "memory");
}

__device__ __forceinline__ void wait_async0() {
  asm volatile("s_wait_asynccnt 0" ::: "memory");
}

__global__ __launch_bounds__(32)
void pcen_wmma_kernel(const float* __restrict__ x,
                      const float* __restrict__ g_log_s,
                      const float* __restrict__ g_log_alpha,
                      const float* __restrict__ g_log_delta,
                      const float* __restrict__ g_log_r,
                      float* __restrict__ out)
{
  __shared__ __align__(16) float sm[2 * BUF_F];

  const int l  = threadIdx.x;        // 0..31 (wave32)
  const int lm = l & 15;
  const int hi = l >> 4;             // half-wave selector

  const int b  = blockIdx.x >> 3;
  const int m0 = (blockIdx.x & 7) << 4;

  const size_t grp = (size_t)(b * M_LEN + m0) * (size_t)T_LEN;
  const float* gx = x + grp;
  float*       gy = out + grp;

  // ---- kick off tile 0 async loads immediately (hide behind setup math) ----
  {
    const unsigned lb = (unsigned)(size_t)&sm[0];
    #pragma unroll
    for (int it = 0; it < ROWS; ++it) {
      const float* g = gx + (size_t)it * T_LEN + 4 * l;
      async_ld_b128(lb + (unsigned)((it * PAD + 4 * l) * 4), (const void*)g);
    }
  }

  const float s     = expf(g_log_s[0]);
  const float alpha = expf(g_log_alpha[0]);
  const float delta = expf(g_log_delta[0]);
  const float r     = expf(g_log_r[0]);
  const float a     = 1.0f - s;
  const float dr    = pw_cold(delta, r);

  // Uniform fast-path flag in an SGPR -> scalar branches, EXEC untouched.
  const int fastAR =
      __builtin_amdgcn_readfirstlane((alpha == 1.0f && r == 1.0f) ? 1 : 0);

  // carry = x[b, m, 0] makes Mema[0] = a*x0 + s*x0 = x0 (exact when s==1).
  float carry = gx[(size_t)lm * T_LEN];

#if HAVE_WMMA_F32X4
  // A operand = L (16x16 lower-triangular, L[i][j] = s*a^(i-j)), split into
  // four 16x4 column blocks. A layout: lane<16 holds (M=lm, K=4k+{0,1});
  // lane>=16 holds (M=lm, K=4k+{2,3}).
  v2f Af[4];
  #pragma unroll
  for (int k = 0; k < 4; ++k) {
    const int j0 = 4 * k + 2 * hi;
    Af[k].x = (j0     <= lm) ? s * ipow(a, lm - j0)     : 0.0f;
    Af[k].y = (j0 + 1 <= lm) ? s * ipow(a, lm - j0 - 1) : 0.0f;
  }
  // Carry weights a^(M+1) for this lane's D rows (M = v + 8*hi).
  float apw[8];
  #pragma unroll
  for (int v = 0; v < 8; ++v) apw[v] = ipow(a, v + 1 + 8 * hi);
#endif

  for (int ti = 0; ti < NTILES; ++ti) {
    const int cur = ti & 1;
    wait_async0();                       // current tile resident in LDS

    // Prefetch next tile into the other buffer (overlaps with compute).
    if (ti + 1 < NTILES) {
      const unsigned lb = (unsigned)(size_t)&sm[(1 - cur) * BUF_F];
      const int t0n = (ti + 1) * TT;
      #pragma unroll
      for (int it = 0; it < ROWS; ++it) {
        const float* g = gx + (size_t)it * T_LEN + t0n + 4 * l;
        async_ld_b128(lb + (unsigned)((it * PAD + 4 * l) * 4), (const void*)g);
      }
    }

    float* tb = &sm[cur * BUF_F];        // tile layout: [m][t], stride PAD

    #pragma unroll
    for (int c = 0; c < TT / 16; ++c) {
      const int tc = c * 16;
#if HAVE_WMMA_F32X4
      // B operand = X chunk (K=t rows x N=m cols); VGPR0 rows {4k,4k+2},
      // VGPR1 rows {4k+1,4k+3}; col = lane%16.
      const float* bp = tb + lm * PAD + tc + 2 * hi;
      const v2f b0 = *(const v2f*)(bp + 0);
      const v2f b1 = *(const v2f*)(bp + 4);
      const v2f b2 = *(const v2f*)(bp + 8);
      const v2f b3 = *(const v2f*)(bp + 12);

      v8f acc;                            // seed with rank-1 carry term
      #pragma unroll
      for (int v = 0; v < 8; ++v) acc[v] = apw[v] * carry;

      acc = __builtin_amdgcn_wmma_f32_16x16x4_f32(false, Af[0], false, b0, (short)0, acc, false, false);
      acc = __builtin_amdgcn_wmma_f32_16x16x4_f32(false, Af[1], false, b1, (short)0, acc, false, false);
      acc = __builtin_amdgcn_wmma_f32_16x16x4_f32(false, Af[2], false, b2, (short)0, acc, false, false);
      acc = __builtin_amdgcn_wmma_f32_16x16x4_f32(false, Af[3], false, b3, (short)0, acc, false, false);

      // New carry = Mema row 15 = D VGPR7 on lanes 16..31; broadcast per column.
      carry = __shfl(acc[7], 16 + lm, 32);

      // pcen elementwise in D layout; overwrite x in place in LDS.
      float* xp = tb + lm * PAD + tc + 8 * hi;
      if (fastAR) {                      // scalar branch (SGPR condition)
        #pragma unroll
        for (int v = 0; v < 8; ++v) {
          const float xv = xp[v];
          const float me = acc[v] + EPS_F;
          const float t  = xv * __builtin_amdgcn_rcpf(me) + delta;
          xp[v] = t - dr;
        }
      } else {
        #pragma unroll
        for (int v = 0; v < 8; ++v) {
          const float xv = xp[v];
          const float me = acc[v] + EPS_F;
          const float dp = pw_hw(me, alpha);                 // me^alpha
          const float t  = xv * __builtin_amdgcn_rcpf(dp) + delta;
          xp[v] = pw_hw(t, r) - dr;
        }
      }
#else
      // VALU fallback: 16 lanes each scan one m-row serially.
      if (hi == 0) {
        float cc = carry;
        float* xp = tb + lm * PAD + tc;
        #pragma unroll
        for (int t = 0; t < 16; ++t) {
          const float xv = xp[t];
          cc = a * cc + s * xv;
          const float me = cc + EPS_F;
          const float dp = (alpha == 1.0f) ? me : pw_hw(me, alpha);
          const float u  = xv * __builtin_amdgcn_rcpf(dp) + delta;
          xp[t] = ((r == 1.0f) ? u : pw_hw(u, r)) - dr;
        }
        carry = cc;
      }
#endif
    }

    // Coalesced store of the finished tile (plain stores: LDS reuse only
    // depends on DScnt, so the next async loads never wait on them).
    const int t0 = ti * TT;
    #pragma unroll
    for (int it = 0; it < ROWS; ++it) {
      const float4 vv = *(const float4*)(tb + it * PAD + 4 * l);
      *(float4*)(gy + (size_t)it * T_LEN + t0 + 4 * l) = vv;
    }
  }
}

extern "C" void kernel_launch(void* const* d_in, const int* in_sizes, int n_in,
                              void* d_out, int out_size, void* d_ws, size_t ws_size,
                              hipStream_t stream) {
  (void)n_in; (void)out_size; (void)d_ws; (void)ws_size;
  const float* x  = (const float*)d_in[0];
  const float* ls = (const float*)d_in[1];
  const float* la = (const float*)d_in[2];
  const float* ld = (const float*)d_in[3];
  const float* lr = (const float*)d_in[4];
  float* o = (float*)d_out;

  const int Bn = in_sizes[0] / (M_LEN * T_LEN);   // 64
  dim3 grid(Bn * (M_LEN / ROWS));                 // 512 one-wave workgroups
  dim3 block(32);
  pcen_wmma_kernel<<<grid, block, 0, stream>>>(x, ls, la, ld, lr, o);
}